// APPNPModule_83459804496278
// MI455X (gfx1250) — compile-verified
//
#include <hip/hip_runtime.h>

typedef __attribute__((ext_vector_type(2))) float v2f;
typedef __attribute__((ext_vector_type(8))) float v8f;

#define ALPHA_F 0.1f

// ---------------------------------------------------------------- utilities
__global__ void zero_i32(int* __restrict__ p, int n) {
  int i = blockIdx.x * blockDim.x + threadIdx.x;
  if (i < n) p[i] = 0;
}

// ---------------------------------------------------- degree count (atomics)
__global__ void degree_kernel(const int* __restrict__ src, const int* __restrict__ dst,
                              int* __restrict__ deg_out, int* __restrict__ deg_in, int E) {
  int e = blockIdx.x * blockDim.x + threadIdx.x;
  if (e < E) {
    atomicAdd(&deg_out[src[e]], 1);
    atomicAdd(&deg_in[dst[e]], 1);
  }
}

// ------------------------------------------------------------- norm factors
__global__ void norm_kernel(const int* __restrict__ deg_out, const int* __restrict__ deg_in,
                            float* __restrict__ ns, float* __restrict__ nd, int N) {
  int i = blockIdx.x * blockDim.x + threadIdx.x;
  if (i < N) {
    int a = deg_out[i], c = deg_in[i];
    ns[i] = (a > 0) ? rsqrtf((float)a) : 0.0f;
    nd[i] = (c > 0) ? rsqrtf((float)c) : 0.0f;
  }
}

// ------------------------------- exclusive scan of deg_in -> row_ptr (1 block)
__global__ void scan_kernel(const int* __restrict__ deg, int* __restrict__ rp, int n) {
  __shared__ int buf[1024];
  __shared__ int carry_s;
  const int tid = threadIdx.x;
  if (tid == 0) carry_s = 0;
  __syncthreads();
  for (int base = 0; base < n; base += 1024) {
    const int i = base + tid;
    const int v = (i < n) ? deg[i] : 0;
    buf[tid] = v;
    __syncthreads();
    for (int off = 1; off < 1024; off <<= 1) {
      int t = 0;
      if (tid >= off) t = buf[tid - off];
      __syncthreads();
      buf[tid] += t;
      __syncthreads();
    }
    const int incl  = buf[tid];
    const int carry = carry_s;
    if (i < n) rp[i] = carry + incl - v;     // exclusive
    const int total = buf[1023];
    __syncthreads();
    if (tid == 0) carry_s = carry + total;
    __syncthreads();
  }
  if (tid == 0) rp[n] = carry_s;
}

// ------------------------------------ CSR-by-destination build + edge weight
__global__ void csr_kernel(const int* __restrict__ src, const int* __restrict__ dst,
                           const int* __restrict__ rp, int* __restrict__ cursor,
                           const float* __restrict__ ns, const float* __restrict__ nd,
                           int* __restrict__ csr_src, float* __restrict__ csr_w, int E) {
  int e = blockIdx.x * blockDim.x + threadIdx.x;
  if (e < E) {
    const int d = dst[e];
    const int s = src[e];
    const int pos = atomicAdd(&cursor[d], 1);
    const int idx = rp[d] + pos;
    csr_src[idx] = s;
    csr_w[idx]   = ns[s] * nd[d];
  }
}

// --------------------------- h0 = x @ W + b via V_WMMA_F32_16X16X4_F32 (fp32)
// One wave computes a 16-row tile x all 64 output columns.
__global__ void gemm_wmma(const float* __restrict__ x, const float* __restrict__ W,
                          const float* __restrict__ bias, float* __restrict__ h0, int N) {
  const int lane = threadIdx.x & 31;
  const int wave = threadIdx.x >> 5;
  const int tile = blockIdx.x * (blockDim.x >> 5) + wave;
  const int ntiles = (N + 15) >> 4;
  if (tile >= ntiles) return;              // uniform per wave; EXEC stays all-1s
  const int m0 = tile << 4;
  const int lm = lane & 15;                // M (for A) / N (for B,C,D) within tile
  const int kh = lane >> 4;                // 0 -> K pair {k,k+1}, 1 -> {k+2,k+3}

  v8f acc0 = {}, acc1 = {}, acc2 = {}, acc3 = {};
  const int rA = (m0 + lm < N) ? (m0 + lm) : (N - 1);
  const float* xrow = x + (size_t)rA * 64;

  for (int k = 0; k < 64; k += 4) {
    const int kk = k + kh * 2;
    v2f a;
    a.x = xrow[kk];
    a.y = xrow[kk + 1];
    const float* w0 = W + (size_t)kk * 64 + lm;
    const float* w1 = W + (size_t)(kk + 1) * 64 + lm;
    v2f b0, b1, b2, b3;
    b0.x = w0[0];   b0.y = w1[0];
    b1.x = w0[16];  b1.y = w1[16];
    b2.x = w0[32];  b2.y = w1[32];
    b3.x = w0[48];  b3.y = w1[48];
    acc0 = __builtin_amdgcn_wmma_f32_16x16x4_f32(false, a, false, b0, (short)0, acc0, false, false);
    acc1 = __builtin_amdgcn_wmma_f32_16x16x4_f32(false, a, false, b1, (short)0, acc1, false, false);
    acc2 = __builtin_amdgcn_wmma_f32_16x16x4_f32(false, a, false, b2, (short)0, acc2, false, false);
    acc3 = __builtin_amdgcn_wmma_f32_16x16x4_f32(false, a, false, b3, (short)0, acc3, false, false);
  }

  const float bb0 = bias[lm], bb1 = bias[16 + lm], bb2 = bias[32 + lm], bb3 = bias[48 + lm];
#pragma unroll
  for (int r = 0; r < 8; ++r) {
    const int row = m0 + r + kh * 8;       // C/D layout: VGPR r -> M=r (lanes 0-15), M=r+8 (16-31)
    if (row < N) {
      float* o = h0 + (size_t)row * 64;
      o[lm]      = acc0[r] + bb0;
      o[16 + lm] = acc1[r] + bb1;
      o[32 + lm] = acc2[r] + bb2;
      o[48 + lm] = acc3[r] + bb3;
    }
  }
}

// --------------- one APPNP round: h_out[d] = 0.9 * sum_e w_e*h_in[src_e] + 0.1*h0[d]
// One wave per node; each lane owns 2 feature columns (float2). Edge (src,w)
// pairs are loaded one per lane (32 at a time) and broadcast with __shfl.
__global__ void prop_kernel(const float* __restrict__ h_in, const float* __restrict__ h0,
                            const int* __restrict__ rp, const int* __restrict__ csr_src,
                            const float* __restrict__ csr_w, float* __restrict__ h_out, int N) {
  const int node = blockIdx.x * (blockDim.x >> 5) + (threadIdx.x >> 5);
  if (node >= N) return;
  const int lane = threadIdx.x & 31;
  const int beg = rp[node];
  const int end = rp[node + 1];

  const v2f* __restrict__ hin2 = (const v2f*)h_in;
  float ax = 0.0f, ay = 0.0f;

  for (int p0 = beg; p0 < end; p0 += 32) {
    const int cnt = end - p0 < 32 ? end - p0 : 32;
    int s = 0; float w = 0.0f;
    if (lane < cnt) {
      s = csr_src[p0 + lane];
      w = csr_w[p0 + lane];
    }
    for (int j = 0; j < cnt; ++j) {
      const int   ss = __shfl(s, j);
      const float ww = __shfl(w, j);
      const v2f v = hin2[(size_t)ss * 32 + lane];   // 256B coalesced per row
      ax += ww * v.x;
      ay += ww * v.y;
    }
  }

  const v2f base = ((const v2f*)h0)[(size_t)node * 32 + lane];
  v2f o;
  o.x = (1.0f - ALPHA_F) * ax + ALPHA_F * base.x;
  o.y = (1.0f - ALPHA_F) * ay + ALPHA_F * base.y;
  ((v2f*)h_out)[(size_t)node * 32 + lane] = o;
}

// --------------------------------------------------------------------- launch
extern "C" void kernel_launch(void* const* d_in, const int* in_sizes, int n_in,
                              void* d_out, int out_size, void* d_ws, size_t ws_size,
                              hipStream_t stream) {
  const float* x    = (const float*)d_in[0];
  const float* W    = (const float*)d_in[1];
  const float* bias = (const float*)d_in[2];
  const int*   src  = (const int*)d_in[3];
  const int*   dst  = (const int*)d_in[4];
  float* out = (float*)d_out;

  const int N = in_sizes[0] / 64;
  const int E = in_sizes[3];

  // carve workspace (256B-aligned slices)
  char* ws = (char*)d_ws;
  auto carve = [&](size_t bytes) -> char* {
    char* p = ws;
    ws += (bytes + 255) & ~(size_t)255;
    return p;
  };
  float* h0      = (float*)carve((size_t)N * 64 * sizeof(float));
  float* hbuf    = (float*)carve((size_t)N * 64 * sizeof(float));
  float* ns      = (float*)carve((size_t)N * sizeof(float));
  float* nd      = (float*)carve((size_t)N * sizeof(float));
  int*   deg_out = (int*)carve((size_t)N * sizeof(int));
  int*   deg_in  = (int*)carve((size_t)N * sizeof(int));
  int*   cursor  = (int*)carve((size_t)N * sizeof(int));
  int*   rp      = (int*)carve((size_t)(N + 1) * sizeof(int));
  int*   csr_src = (int*)carve((size_t)E * sizeof(int));
  float* csr_w   = (float*)carve((size_t)E * sizeof(float));

  const int TB = 256;
  zero_i32<<<(N + TB - 1) / TB, TB, 0, stream>>>(deg_out, N);
  zero_i32<<<(N + TB - 1) / TB, TB, 0, stream>>>(deg_in, N);
  zero_i32<<<(N + TB - 1) / TB, TB, 0, stream>>>(cursor, N);

  degree_kernel<<<(E + TB - 1) / TB, TB, 0, stream>>>(src, dst, deg_out, deg_in, E);
  norm_kernel<<<(N + TB - 1) / TB, TB, 0, stream>>>(deg_out, deg_in, ns, nd, N);
  scan_kernel<<<1, 1024, 0, stream>>>(deg_in, rp, N);
  csr_kernel<<<(E + TB - 1) / TB, TB, 0, stream>>>(src, dst, rp, cursor, ns, nd, csr_src, csr_w, E);

  const int ntiles = (N + 15) / 16;
  gemm_wmma<<<(ntiles + 3) / 4, 128, 0, stream>>>(x, W, bias, h0, N);

  // K=10 rounds; odd rounds write hbuf, even rounds write d_out (round 10 -> d_out)
  const int nodeBlocks = (N + 7) / 8;   // 8 waves (nodes) per 256-thread block
  const float* cur = h0;
  for (int k = 1; k <= 10; ++k) {
    float* o = (k & 1) ? hbuf : out;
    prop_kernel<<<nodeBlocks, TB, 0, stream>>>(cur, h0, rp, csr_src, csr_w, o, N);
    cur = o;
  }
}